// ResCostVolume_13718125543602
// MI455X (gfx1250) — compile-verified
//
#include <hip/hip_runtime.h>
#include <stdint.h>

// Residual L1 cost volume for MI455X (gfx1250, wave32).
//   out[n,d,h,w] = sum_c |x[n,c,h,w] - y[n,c,h,w-(d-12)]|, 0 outside overlap.
// Strategy: 1 workgroup per (n,h). Stage the strided y row (32x512 f32, 64KB)
// into LDS with the Tensor Data Mover (2D tile descriptor, TENSORcnt-tracked),
// stream x straight from HBM (each element read exactly once, reused across
// all 25 disparities in registers). ~356MB total HBM traffic => ~15us floor.

#define MAXDISP 25
#define DOFF    12          // (MAXDISP-1)/2
#define CCH     32
#define HH      256
#define WW      512
#define PAD     16          // front/back LDS padding so shifted windows stay in-bounds

typedef __attribute__((ext_vector_type(4))) unsigned int v4u;
typedef __attribute__((ext_vector_type(8))) int          v8i;
typedef __attribute__((ext_vector_type(4))) int          v4i;

__global__ __launch_bounds__(256) void
rescost_tdm_kernel(const float* __restrict__ x,
                   const float* __restrict__ y,
                   float* __restrict__ out) {
    // y row staging buffer with padding so reads at w0-12 .. w0+13 never
    // leave the array (out-of-overlap lanes read garbage that is masked off
    // at store time).
    __shared__ float ysbuf[PAD + CCH * WW + PAD];
    float* ys = ysbuf + PAD;

    const int nh  = blockIdx.x;
    const int n   = nh >> 8;          // / HH
    const int h   = nh & (HH - 1);
    const int tid = threadIdx.x;

    // ---- TDM: stage y[n, 0:32, h, 0:512] (2D tile, stride H*W between rows)
    // into LDS. Only wave 0 issues it (TDM ignores EXEC, so gate at wave
    // granularity with a wave-uniform branch).
    if ((tid >> 5) == 0) {
        const uint64_t gaddr =
            (uint64_t)(uintptr_t)(y + ((((size_t)n * CCH) * HH + h) * WW));
        const uint32_t lds_off = (uint32_t)(uintptr_t)(void*)ys;

        // D# group 0 (128b): count=1 | lds_addr | global_addr[56:0] | type=2
        v4u g0;
        g0.x = 1u;                                          // count=1, user mode
        g0.y = lds_off;                                     // lds_addr (bytes)
        g0.z = (uint32_t)(gaddr & 0xFFFFFFFFu);             // global_addr[31:0]
        g0.w = (uint32_t)((gaddr >> 32) & 0x01FFFFFFu)      // global_addr[56:32]
             | (2u << 30);                                  // type = 2 ("image")

        // D# group 1 (256b): data_size=4B, tensor 512x32, tile 512x32,
        // tensor_dim0_stride = H*W elements.
        v8i g1;
        g1[0] = 0x00020000;                 // wg_mask=0 | data_size=2 (4B) | no flags
        g1[1] = (int)((WW & 0xFFFF) << 16); // barrier_addr=0 | tensor_dim0[15:0]
        g1[2] = (int)((WW >> 16) | ((CCH & 0xFFFF) << 16)); // dim0 hi | dim1 lo
        g1[3] = (int)(((CCH >> 16) & 0xFFFF) | (WW << 16)); // dim1 hi | tile_dim0=512
        g1[4] = CCH;                        // tile_dim1=32 | tile_dim2=0
        g1[5] = HH * WW;                    // tensor_dim0_stride[31:0] = 131072
        g1[6] = 0;                          // stride hi | tensor_dim1_stride lo
        g1[7] = 0;                          // tensor_dim1_stride hi

        v4i g2 = {0, 0, 0, 0};              // <=2D tensor: groups 2/3 unused
        v4i g3 = {0, 0, 0, 0};
        v8i g4 = {0, 0, 0, 0, 0, 0, 0, 0};  // extra descriptor group (unused, 2D)
        __builtin_amdgcn_tensor_load_to_lds(g0, g1, g2, g3, g4, 0);
        __builtin_amdgcn_s_wait_tensorcnt(0);   // s_wait_tensorcnt 0
    }
    __syncthreads();

    // ---- Compute: each thread owns columns (w0, w0+1) for all 25 disparities.
    const int w0 = tid << 1;

    float acc[MAXDISP][2];
#pragma unroll
    for (int d = 0; d < MAXDISP; ++d) { acc[d][0] = 0.f; acc[d][1] = 0.f; }

    const float* xcol = x + ((((size_t)n * CCH) * HH + h) * WW) + w0;

    for (int c = 0; c < CCH; ++c) {
        // x streamed once from HBM, reused across all 25 disparities in regs.
        const float2 xv = *(const float2*)(xcol + (size_t)c * (HH * WW));

        // 26-wide y window from LDS: y[c][w0-12 .. w0+13].
        // 8B-aligned (w0 even) -> ds_load_b64 pairs, conflict-free banks.
        const float* yc = ys + c * WW + (w0 - DOFF);
        float yv[26];
#pragma unroll
        for (int j = 0; j < 26; ++j) yv[j] = yc[j];

#pragma unroll
        for (int d = 0; d < MAXDISP; ++d) {
            // y index for column w is w + DOFF - d  ->  window slot 24-d / 25-d
            acc[d][0] += fabsf(xv.x - yv[24 - d]);
            acc[d][1] += fabsf(xv.y - yv[25 - d]);
        }
    }

    // ---- Store with overlap masking (exact zeros outside the valid region).
    float* orow = out + ((((size_t)n * MAXDISP) * HH + h) * WW) + w0;
#pragma unroll
    for (int d = 0; d < MAXDISP; ++d) {
        const int dp  = d - DOFF;
        const int yi0 = w0 - dp;
        const int yi1 = w0 + 1 - dp;
        float2 r;
        r.x = (yi0 >= 0 && yi0 < WW) ? acc[d][0] : 0.f;
        r.y = (yi1 >= 0 && yi1 < WW) ? acc[d][1] : 0.f;
        *(float2*)(orow + (size_t)d * (HH * WW)) = r;
    }
}

extern "C" void kernel_launch(void* const* d_in, const int* in_sizes, int n_in,
                              void* d_out, int out_size, void* d_ws, size_t ws_size,
                              hipStream_t stream) {
    (void)in_sizes; (void)n_in; (void)d_ws; (void)ws_size; (void)out_size;
    const float* x = (const float*)d_in[0];
    const float* y = (const float*)d_in[1];
    float* out     = (float*)d_out;

    const int nbatch = 8;
    dim3 grid(nbatch * HH);     // one workgroup per (n, h) row: 2048 WGs
    dim3 block(256);            // 8 wave32 waves
    rescost_tdm_kernel<<<grid, block, 0, stream>>>(x, y, out);
}